// OneMatrixFunction_71871982731802
// MI455X (gfx1250) — compile-verified
//
#include <hip/hip_runtime.h>
#include <math.h>

typedef float v2f __attribute__((ext_vector_type(2)));
typedef float v8f __attribute__((ext_vector_type(8)));

#define DD      16384            // SUM_DIMS
#define HB      4096             // block size (H0)
#define BATCH   8
#define KSPLIT  8
#define KCHUNK  (HB / KSPLIT)    // 512
#define TILE_N  64
#define TILES_PER_BLK (HB / TILE_N) // 64
#define NPAIR   6
#define NSLOT   (2 * KSPLIT)     // 16 deterministic partial slots

// workspace layout in floats
#define OFF_RHOT 0                           // [DD][16]  (A fragments, rows 8..15 zero)
#define OFF_RHOD (DD * 16)                   // [BATCH][DD]
#define OFF_ACC  (OFF_RHOD + BATCH * DD)     // [NSLOT][BATCH][DD]
#define ACC_FLOATS (NSLOT * BATCH * DD)

// ---------------------------------------------------------------- prep ------
// 16384 threads: thread k builds rho_T column k (16 floats, padded), rho',
// and zeroes the partial-accumulator slots with b128 stores.
__global__ void __launch_bounds__(256)
prep_kernel(const float* __restrict__ s, float* __restrict__ ws) {
  float* rhoT = ws + OFF_RHOT;
  float* rhod = ws + OFF_RHOD;
  float4* acc4 = (float4*)(ws + OFF_ACC);
  const int k = blockIdx.x * blockDim.x + threadIdx.x; // 0..DD-1
#pragma unroll
  for (int m = 0; m < BATCH; ++m) {
    float x = s[(size_t)m * DD + k];
    float r = 1.0f / (1.0f + expf(-4.0f * (x - 0.5f)));
    rhoT[(size_t)k * 16 + m] = r;
    rhod[(size_t)m * DD + k] = 4.0f * r * (1.0f - r);
  }
#pragma unroll
  for (int m = BATCH; m < 16; ++m) rhoT[(size_t)k * 16 + m] = 0.0f;
  const float4 z = make_float4(0.f, 0.f, 0.f, 0.f);
  for (int i = k; i < ACC_FLOATS / 4; i += DD) acc4[i] = z;
}

// ---------------------------------------------------------------- gemm ------
// One wave per (block-pair, 64-col tile, K chunk). 4 WMMA accumulator chains
// per wave (16x64 padded-M tile), f32 WMMA 16x16x4. W read with b128 loads.
__global__ void __launch_bounds__(256)
gemm_kernel(const float* __restrict__ W, const float* __restrict__ rhoT,
            float* __restrict__ acc) {
  const int wave = (blockIdx.x * blockDim.x + threadIdx.x) >> 5; // 0..3071
  const int lane = threadIdx.x & 31;
  const int l    = lane & 15;
  const int half = lane >> 4;

  const int pair = wave / (TILES_PER_BLK * KSPLIT);     // 0..5
  const int rem  = wave % (TILES_PER_BLK * KSPLIT);
  const int tile = rem / KSPLIT;
  const int kch  = rem % KSPLIT;

  const int dst = (pair + 1) >> 1;                      // 0,1,1,2,2,3
  const int src = (pair & 1) ? (dst - 1) : (dst + 1);   // 1,0,2,1,3,2
  const int sel = ((pair & 1) == 0 && pair >= 2) ? 1 : 0;

  const int j0 = dst * HB + tile * TILE_N;
  const int k0 = src * HB + kch * KCHUNK;

  // W fragment pointers: lanes 0-15 read rows k,k+1; lanes 16-31 rows k+2,k+3
  const float4* p0 = (const float4*)(W + (size_t)(k0 + 2 * half) * DD + j0) + l;
  const float4* p1 = p0 + (DD / 4);              // +1 row
  // A fragment pointer into rho_T (matches 32-bit A 16x4 VGPR layout)
  const float* pa = rhoT + (size_t)k0 * 16 + (lane < 16 ? lane : lane + 16);

  v8f c0 = {}, c1 = {}, c2 = {}, c3 = {};

#pragma unroll 4
  for (int it = 0; it < KCHUNK / 4; ++it) {
    const float4 w0 = *p0;                       // row k+2h  , cols j0+4l..+3
    const float4 w1 = *p1;                       // row k+2h+1
    const float a0 = pa[0];                      // A K even
    const float a1 = pa[16];                     // A K odd
    p0 += DD;                                    // advance 4 rows (float4 units)
    p1 += DD;
    pa += 64;

    v2f a  = {a0, a1};
    v2f b0 = {w0.x, w1.x};
    v2f b1 = {w0.y, w1.y};
    v2f b2 = {w0.z, w1.z};
    v2f b3 = {w0.w, w1.w};

    c0 = __builtin_amdgcn_wmma_f32_16x16x4_f32(false, a, false, b0, (short)0, c0, false, false);
    c1 = __builtin_amdgcn_wmma_f32_16x16x4_f32(false, a, false, b1, (short)0, c1, false, false);
    c2 = __builtin_amdgcn_wmma_f32_16x16x4_f32(false, a, false, b2, (short)0, c2, false, false);
    c3 = __builtin_amdgcn_wmma_f32_16x16x4_f32(false, a, false, b3, (short)0, c3, false, false);
  }

  // C layout: VGPR r, lanes 0-15 hold M=r, N=lane. Batch rows are 0..7.
  // Sub-tile s holds logical column j0 + 4*n + s  ->  float4 store per row.
  if (lane < 16) {
    float* outp = acc + (size_t)(sel * KSPLIT + kch) * (BATCH * DD) + j0 + 4 * l;
#pragma unroll
    for (int r = 0; r < BATCH; ++r) {
      float4 v = make_float4(c0[r], c1[r], c2[r], c3[r]);
      *(float4*)(outp + (size_t)r * DD) = v;
    }
  }
}

// ------------------------------------------------------------- epilogue -----
__global__ void __launch_bounds__(256)
epi_kernel(const float* __restrict__ acc, const float* __restrict__ rhod,
           const float* __restrict__ Ux, const float* __restrict__ bias,
           float* __restrict__ out) {
  const int tid = blockIdx.x * blockDim.x + threadIdx.x; // 0..BATCH*DD-1
  const int b = tid >> 14;          // /DD
  const int j = tid & (DD - 1);
  float sum = 0.0f;
#pragma unroll
  for (int sl = 0; sl < NSLOT; ++sl) sum += acc[(size_t)sl * (BATCH * DD) + tid];
  sum += bias[j];
  if (j < HB) sum += Ux[b * HB + j];
  out[tid] = sum * rhod[tid];
}

// ---------------------------------------------------------------------------
extern "C" void kernel_launch(void* const* d_in, const int* in_sizes, int n_in,
                              void* d_out, int out_size, void* d_ws, size_t ws_size,
                              hipStream_t stream) {
  (void)in_sizes; (void)n_in; (void)out_size; (void)ws_size;
  const float* Ux   = (const float*)d_in[0];   // [8, 4096]
  const float* s    = (const float*)d_in[1];   // [8, 16384]
  const float* W    = (const float*)d_in[2];   // [16384, 16384]
  const float* bias = (const float*)d_in[3];   // [1, 16384]
  float* out = (float*)d_out;                  // [8, 16384]
  float* ws  = (float*)d_ws;

  prep_kernel<<<DD / 256, 256, 0, stream>>>(s, ws);

  const int gemm_waves  = NPAIR * TILES_PER_BLK * KSPLIT;   // 3072
  const int gemm_blocks = gemm_waves * 32 / 256;            // 384
  gemm_kernel<<<gemm_blocks, 256, 0, stream>>>(W, ws + OFF_RHOT, ws + OFF_ACC);

  epi_kernel<<<(BATCH * DD) / 256, 256, 0, stream>>>(ws + OFF_ACC, ws + OFF_RHOD,
                                                     Ux, bias, out);
}